// Explainer_16767552323790
// MI455X (gfx1250) — compile-verified
//
#include <hip/hip_runtime.h>
#include <hip/hip_bf16.h>

// ---------------------------------------------------------------------------
// GINE-style GNN (2 message-passing blocks + MLP head) for MI455X / gfx1250.
// All GEMMs run on the matrix pipe via V_WMMA_F32_16X16X4_F32 (fp32 in/out,
// matches reference precision). The workload is gather/atomic bound (~2.5 GB
// of traffic -> ~110us at 23.3 TB/s), so edge messages are fused with the
// scatter-add and never materialized in HBM.
// ---------------------------------------------------------------------------

typedef __attribute__((ext_vector_type(2))) float v2f;
typedef __attribute__((ext_vector_type(8))) float v8f;

#define WMMA_F32(a, b, c) \
    __builtin_amdgcn_wmma_f32_16x16x4_f32(false, (a), false, (b), (short)0, (c), false, false)

static constexpr int N_NODES = 100000;
static constexpr int N_EDGES = 1600000;
static constexpr int HID     = 128;

// ---------------------------------------------------------------------------
// Utility: fill
// ---------------------------------------------------------------------------
__global__ void fill_f32(float* __restrict__ p, float v, size_t n) {
    size_t i  = (size_t)blockIdx.x * blockDim.x + threadIdx.x;
    size_t st = (size_t)gridDim.x * blockDim.x;
    for (; i < n; i += st) p[i] = v;
}

// out = a + b (elementwise)
__global__ void vadd_f32(const float* __restrict__ a, const float* __restrict__ b,
                         float* __restrict__ o, size_t n) {
    size_t i  = (size_t)blockIdx.x * blockDim.x + threadIdx.x;
    size_t st = (size_t)gridDim.x * blockDim.x;
    for (; i < n; i += st) o[i] = a[i] + b[i];
}

// ---------------------------------------------------------------------------
// Edge kernel: msg = relu(x[src] + e @ We + be); agg[dst] += msg
// One wave per tile of 16 edges. A = edge_attr tile [16 x 16] (K=16 via 4
// chained 16x16x4 WMMAs), B = We column tile [16 x 16]. D = 32 or 128.
// ---------------------------------------------------------------------------
template <int D>
__global__ void edge_msg_scatter(const float* __restrict__ x,     // [N, D]
                                 const int*   __restrict__ src,   // [E]
                                 const int*   __restrict__ dst,   // [E]
                                 const float* __restrict__ ea,    // [E, 16]
                                 const float* __restrict__ We,    // [16, D]
                                 const float* __restrict__ be,    // [D]
                                 float* __restrict__ agg,         // [N, D]
                                 int nTiles) {
    int gw = (int)((blockIdx.x * blockDim.x + threadIdx.x) >> 5);
    if (gw >= nTiles) return;
    int lane = threadIdx.x & 31;
    int half = lane >> 4;          // 0: lanes 0-15, 1: lanes 16-31
    int col  = lane & 15;
    int e0   = gw * 16;

    // A fragments: lane holds row M=col; VGPR pair = K {kb, kb+1},
    // kb = 4*kk + 2*half per the 32-bit 16x4 A layout.
    const float* arow = ea + (size_t)(e0 + col) * 16 + half * 2;
    v2f afrag[4];
#pragma unroll
    for (int kk = 0; kk < 4; ++kk)
        afrag[kk] = *(const v2f*)(arow + kk * 4);

    // Rows of the C tile owned by this lane: M = half*8 + i
    int srows[8], drows[8];
#pragma unroll
    for (int i = 0; i < 8; ++i) {
        int m    = e0 + half * 8 + i;
        srows[i] = src[m];
        drows[i] = dst[m];
    }

#pragma unroll
    for (int ct = 0; ct < D / 16; ++ct) {
        int ncol = ct * 16 + col;
        v8f acc  = {};
#pragma unroll
        for (int kk = 0; kk < 4; ++kk) {
            int kb = kk * 4 + half * 2;
            v2f b;
            b.x = We[(size_t)kb * D + ncol];
            b.y = We[(size_t)(kb + 1) * D + ncol];
            acc = WMMA_F32(afrag[kk], b, acc);
        }
        float bv = be[ncol];
#pragma unroll
        for (int i = 0; i < 8; ++i) {
            float v = acc[i] + bv + x[(size_t)srows[i] * D + ncol];
            v       = fmaxf(v, 0.0f);
            atomicAdd(agg + (size_t)drows[i] * D + ncol, v);
        }
    }
}

// ---------------------------------------------------------------------------
// Node GEMM: out[N x 128] = act(in[N x DIN] @ W[DIN x 128] + bias)
// Wave owns one 16-wide column tile; preloads all B fragments for the K loop
// into registers (reused across node tiles), grid-strides over node tiles.
// ---------------------------------------------------------------------------
template <int DIN, bool RELU>
__global__ void gemm_node(const float* __restrict__ in, const float* __restrict__ W,
                          const float* __restrict__ bias, float* __restrict__ out,
                          int nrows) {
    const int NT     = nrows >> 4;  // node tiles (nrows % 16 == 0)
    int gw           = (int)((blockIdx.x * blockDim.x + threadIdx.x) >> 5);
    int totalWaves   = (int)((gridDim.x * blockDim.x) >> 5);
    int lane         = threadIdx.x & 31;
    int half         = lane >> 4;
    int col          = lane & 15;
    int ct           = gw & 7;        // 8 column tiles of 16 (HID=128)
    int wct          = gw >> 3;
    int nWct         = totalWaves >> 3;
    int ncol         = ct * 16 + col;

    constexpr int KS = DIN / 4;
    v2f bfrag[KS];
#pragma unroll
    for (int kk = 0; kk < KS; ++kk) {
        int kb      = kk * 4 + half * 2;
        bfrag[kk].x = W[(size_t)kb * HID + ncol];
        bfrag[kk].y = W[(size_t)(kb + 1) * HID + ncol];
    }
    float bv = bias[ncol];

    for (int nt = wct; nt < NT; nt += nWct) {
        const float* arow = in + (size_t)(nt * 16 + col) * DIN + half * 2;
        v8f acc = {};
#pragma unroll
        for (int kk = 0; kk < KS; ++kk) {
            v2f a = *(const v2f*)(arow + kk * 4);
            acc   = WMMA_F32(a, bfrag[kk], acc);
        }
        float* orow = out + (size_t)(nt * 16 + half * 8) * HID + ncol;
#pragma unroll
        for (int i = 0; i < 8; ++i) {
            float v = acc[i] + bv;
            if (RELU) v = fmaxf(v, 0.0f);
            orow[(size_t)i * HID] = v;
        }
    }
}

// ---------------------------------------------------------------------------
// BatchNorm over the node dimension (per-channel), 3 stages.
// ---------------------------------------------------------------------------
__global__ void bn_stats(const float* __restrict__ h, float* __restrict__ stats,
                         int nrows) {
    int   c  = threadIdx.x;  // blockDim.x == 128
    float s  = 0.0f;
    float s2 = 0.0f;
    for (int r = blockIdx.x; r < nrows; r += gridDim.x) {
        float v = h[(size_t)r * HID + c];
        s += v;
        s2 += v * v;
    }
    atomicAdd(&stats[c], s);
    atomicAdd(&stats[HID + c], s2);
}

__global__ void bn_finalize(float* __restrict__ stats, const float* __restrict__ gamma,
                            const float* __restrict__ beta, float inv_n) {
    int   c   = threadIdx.x;  // 128 threads
    float mu  = stats[c] * inv_n;
    float var = stats[HID + c] * inv_n - mu * mu;
    float sc  = gamma[c] * rsqrtf(var + 1e-5f);
    stats[2 * HID + c] = sc;
    stats[3 * HID + c] = beta[c] - mu * sc;
}

__global__ void bn_apply_relu(const float* __restrict__ h, const float* __restrict__ stats,
                              float* __restrict__ o, size_t n) {
    size_t i  = (size_t)blockIdx.x * blockDim.x + threadIdx.x;
    size_t st = (size_t)gridDim.x * blockDim.x;
    for (; i < n; i += st) {
        int c = (int)(i & (HID - 1));
        o[i]  = fmaxf(fmaf(h[i], stats[2 * HID + c], stats[3 * HID + c]), 0.0f);
    }
}

// ---------------------------------------------------------------------------
// Head matvec + sigmoid: out[n] = sigmoid(dot(t[n,:], w) + b). One wave/node.
// ---------------------------------------------------------------------------
__global__ void head_out_kernel(const float* __restrict__ t, const float* __restrict__ w,
                                const float* __restrict__ b2, float* __restrict__ out,
                                int n) {
    int wid  = (int)((blockIdx.x * blockDim.x + threadIdx.x) >> 5);
    int lane = threadIdx.x & 31;
    if (wid >= n) return;
    const float4* row = (const float4*)(t + (size_t)wid * HID);
    const float4* wv  = (const float4*)w;
    float4 a  = row[lane];
    float4 ww = wv[lane];
    float  s  = a.x * ww.x + a.y * ww.y + a.z * ww.z + a.w * ww.w;
#pragma unroll
    for (int off = 16; off > 0; off >>= 1) s += __shfl_xor(s, off, 32);
    if (lane == 0) out[wid] = 1.0f / (1.0f + __expf(-(s + b2[0])));
}

// ---------------------------------------------------------------------------
// Launcher
// ---------------------------------------------------------------------------
extern "C" void kernel_launch(void* const* d_in, const int* in_sizes, int n_in,
                              void* d_out, int out_size, void* d_ws, size_t ws_size,
                              hipStream_t stream) {
    const float* x   = (const float*)d_in[0];
    const int*   ei  = (const int*)d_in[1];
    const float* ea  = (const float*)d_in[2];
    const float* l0_We    = (const float*)d_in[3];
    const float* l0_be    = (const float*)d_in[4];
    const float* l0_W1    = (const float*)d_in[5];
    const float* l0_b1    = (const float*)d_in[6];
    const float* l0_W2    = (const float*)d_in[7];
    const float* l0_b2    = (const float*)d_in[8];
    const float* l0_gamma = (const float*)d_in[9];
    const float* l0_beta  = (const float*)d_in[10];
    const float* l1_We    = (const float*)d_in[11];
    const float* l1_be    = (const float*)d_in[12];
    const float* l1_W1    = (const float*)d_in[13];
    const float* l1_b1    = (const float*)d_in[14];
    const float* l1_W2    = (const float*)d_in[15];
    const float* l1_b2    = (const float*)d_in[16];
    const float* l1_gamma = (const float*)d_in[17];
    const float* l1_beta  = (const float*)d_in[18];
    const float* head_W1  = (const float*)d_in[19];
    const float* head_b1  = (const float*)d_in[20];
    const float* head_W2  = (const float*)d_in[21];
    const float* head_b2  = (const float*)d_in[22];

    const int N = N_NODES, E = N_EDGES;
    const int* srcp = ei;
    const int* dstp = ei + E;

    // workspace layout
    float*       ws    = (float*)d_ws;
    float*       stats = ws;                    // 512 floats
    const size_t FSZ   = (size_t)N * HID;       // 12.8M floats
    float*       A     = ws + 512;
    float*       B     = A + FSZ;
    float*       C     = B + FSZ;

    const int nTiles     = E / 16;              // 100000
    const int edgeBlocks = nTiles / 8;          // 8 waves/block of 256 thr
    const int gemmBlocks = 512;
    const int ewBlocks   = 2048;

    // ---------------- block 0 (DIN = 32) ----------------
    fill_f32<<<512, 256, 0, stream>>>(A, 0.0f, (size_t)N * 32);
    edge_msg_scatter<32><<<edgeBlocks, 256, 0, stream>>>(x, srcp, dstp, ea, l0_We, l0_be, A, nTiles);
    vadd_f32<<<ewBlocks, 256, 0, stream>>>(x, A, B, (size_t)N * 32);
    gemm_node<32, true><<<gemmBlocks, 256, 0, stream>>>(B, l0_W1, l0_b1, A, N);
    gemm_node<128, false><<<gemmBlocks, 256, 0, stream>>>(A, l0_W2, l0_b2, B, N);
    fill_f32<<<1, 256, 0, stream>>>(stats, 0.0f, 256);
    bn_stats<<<512, 128, 0, stream>>>(B, stats, N);
    bn_finalize<<<1, 128, 0, stream>>>(stats, l0_gamma, l0_beta, 1.0f / (float)N);
    bn_apply_relu<<<ewBlocks, 256, 0, stream>>>(B, stats, C, FSZ);

    // ---------------- block 1 (DIN = 128) ----------------
    fill_f32<<<2048, 256, 0, stream>>>(A, 0.0f, FSZ);
    edge_msg_scatter<128><<<edgeBlocks, 256, 0, stream>>>(C, srcp, dstp, ea, l1_We, l1_be, A, nTiles);
    vadd_f32<<<ewBlocks, 256, 0, stream>>>(C, A, B, FSZ);
    gemm_node<128, true><<<gemmBlocks, 256, 0, stream>>>(B, l1_W1, l1_b1, A, N);
    gemm_node<128, false><<<gemmBlocks, 256, 0, stream>>>(A, l1_W2, l1_b2, B, N);
    fill_f32<<<1, 256, 0, stream>>>(stats, 0.0f, 256);
    bn_stats<<<512, 128, 0, stream>>>(B, stats, N);
    bn_finalize<<<1, 128, 0, stream>>>(stats, l1_gamma, l1_beta, 1.0f / (float)N);
    bn_apply_relu<<<ewBlocks, 256, 0, stream>>>(B, stats, C, FSZ);

    // ---------------- head ----------------
    gemm_node<128, true><<<gemmBlocks, 256, 0, stream>>>(C, head_W1, head_b1, A, N);
    head_out_kernel<<<(N + 7) / 8, 256, 0, stream>>>(A, head_W2, head_b2, (float*)d_out, N);
}